// ContrastLoss_12154757447946
// MI455X (gfx1250) — compile-verified
//
#include <hip/hip_runtime.h>

typedef float v2f __attribute__((ext_vector_type(2)));
typedef float v4f __attribute__((ext_vector_type(4)));
typedef float v8f __attribute__((ext_vector_type(8)));

// Sum a per-lane scalar across all 32 lanes of the wave using one
// V_WMMA_F32_16X16X4_F32 with B == ones:
//   D = A x ones  =>  every column of D is the row-sums of A, so
//   (per-lane sum of the 8 C VGPRs) + shfl_xor(16) == sum of all 32 partials.
// EXEC must be all ones here (full blocks, uniform trip counts).
__device__ __forceinline__ float wave_reduce_wmma(float p) {
    v2f a; a[0] = p;    a[1] = 0.0f;
    v2f b; b[0] = 1.0f; b[1] = 1.0f;
    v8f c = {};
    c = __builtin_amdgcn_wmma_f32_16x16x4_f32(
            false, a, false, b, (short)0, c, false, false);
    float r = (c[0] + c[1]) + (c[2] + c[3]) + (c[4] + c[5]) + (c[6] + c[7]);
    r += __shfl_xor(r, 16, 32);   // combine the two half-wave row groups
    return r;
}

__global__ void __launch_bounds__(256)
diag_dot_partials(const float* __restrict__ s,
                  const float* __restrict__ t,
                  float* __restrict__ partials,
                  int n4) {
    const v4f* s4 = (const v4f*)s;
    const v4f* t4 = (const v4f*)t;
    const int S = (int)(gridDim.x * blockDim.x);       // stride in float4 units
    int i = (int)(blockIdx.x * blockDim.x + threadIdx.x);

    // 4 independent accumulators; 8 non-temporal b128 loads in flight per
    // iteration (256 MB single-use footprint > 192 MB L2 -> NT policy).
    float a0 = 0.0f, a1 = 0.0f, a2 = 0.0f, a3 = 0.0f;
    for (; i + 3 * S < n4; i += 4 * S) {
        v4f s0 = __builtin_nontemporal_load(s4 + i);
        v4f s1 = __builtin_nontemporal_load(s4 + i + S);
        v4f s2 = __builtin_nontemporal_load(s4 + i + 2 * S);
        v4f s3 = __builtin_nontemporal_load(s4 + i + 3 * S);
        v4f t0 = __builtin_nontemporal_load(t4 + i);
        v4f t1 = __builtin_nontemporal_load(t4 + i + S);
        v4f t2 = __builtin_nontemporal_load(t4 + i + 2 * S);
        v4f t3 = __builtin_nontemporal_load(t4 + i + 3 * S);
        a0 = fmaf(s0[0], t0[0], a0); a0 = fmaf(s0[1], t0[1], a0);
        a0 = fmaf(s0[2], t0[2], a0); a0 = fmaf(s0[3], t0[3], a0);
        a1 = fmaf(s1[0], t1[0], a1); a1 = fmaf(s1[1], t1[1], a1);
        a1 = fmaf(s1[2], t1[2], a1); a1 = fmaf(s1[3], t1[3], a1);
        a2 = fmaf(s2[0], t2[0], a2); a2 = fmaf(s2[1], t2[1], a2);
        a2 = fmaf(s2[2], t2[2], a2); a2 = fmaf(s2[3], t2[3], a2);
        a3 = fmaf(s3[0], t3[0], a3); a3 = fmaf(s3[1], t3[1], a3);
        a3 = fmaf(s3[2], t3[2], a3); a3 = fmaf(s3[3], t3[3], a3);
    }
    // Remainder (empty for the reference geometry; kept for generality).
    for (; i < n4; i += S) {
        v4f sv = __builtin_nontemporal_load(s4 + i);
        v4f tv = __builtin_nontemporal_load(t4 + i);
        a0 = fmaf(sv[0], tv[0], a0); a0 = fmaf(sv[1], tv[1], a0);
        a0 = fmaf(sv[2], tv[2], a0); a0 = fmaf(sv[3], tv[3], a0);
    }
    float acc = (a0 + a1) + (a2 + a3);

    float wsum = wave_reduce_wmma(acc);

    __shared__ float lds[8];            // 256 threads / wave32 = 8 waves
    const int lane = threadIdx.x & 31;
    const int wave = threadIdx.x >> 5;
    if (lane == 0) lds[wave] = wsum;
    __syncthreads();
    if (threadIdx.x == 0) {
        float bsum = 0.0f;
#pragma unroll
        for (int w = 0; w < 8; ++w) bsum += lds[w];
        partials[blockIdx.x] = bsum;
    }
}

__global__ void __launch_bounds__(256)
final_reduce(const float* __restrict__ partials, int n,
             float* __restrict__ out, float inv_bsz) {
    __shared__ float lds[256];
    float acc = 0.0f;
    for (int i = threadIdx.x; i < n; i += 256) acc += partials[i];
    lds[threadIdx.x] = acc;
    __syncthreads();
    for (int off = 128; off > 0; off >>= 1) {
        if ((int)threadIdx.x < off) lds[threadIdx.x] += lds[threadIdx.x + off];
        __syncthreads();
    }
    if (threadIdx.x == 0) out[0] = -lds[0] * inv_bsz;   // loss = -sum / bsz
}

extern "C" void kernel_launch(void* const* d_in, const int* in_sizes, int n_in,
                              void* d_out, int out_size, void* d_ws, size_t ws_size,
                              hipStream_t stream) {
    const float* f_s = (const float*)d_in[0];
    const float* f_t = (const float*)d_in[1];
    float* out = (float*)d_out;
    float* partials = (float*)d_ws;      // BLOCKS * 4 bytes = 8 KB scratch

    const int n   = in_sizes[0];          // B * D = 33,554,432
    const int n4  = n / 4;                // 8,388,608 float4
    const int bsz = n / 512;              // D = 512 per the reference

    const int THREADS = 256;              // 8 wave32 per block
    const int BLOCKS  = 2048;             // 524,288 threads -> 16 f4/thread

    diag_dot_partials<<<BLOCKS, THREADS, 0, stream>>>(f_s, f_t, partials, n4);
    final_reduce<<<1, THREADS, 0, stream>>>(partials, BLOCKS, out,
                                            1.0f / (float)bsz);
}